// ComputeYoloLoss_47347719471131
// MI455X (gfx1250) — compile-verified
//
#include <hip/hip_runtime.h>
#include <math.h>
#include <stdint.h>

#define DEV __device__ __forceinline__

// ---------------- problem constants ----------------
constexpr int B  = 16;
constexpr int M  = 32;
constexpr int NC = 80;
constexpr int A0 = 6400, A1 = 1600, A2v = 400;
constexpr int A  = A0 + A1 + A2v;          // 8400
constexpr int BA = B * A;                  // 134400
constexpr int BM = B * M;                  // 512
constexpr long long BMA = (long long)BM * A;
constexpr int NBLK = BA / 256;             // 525 (exact)
constexpr int PPAD = 576;                  // 9 tiles of 64 for WMMA reduction
constexpr float EPSA = 1e-9f;
constexpr float INPUT = 640.0f;            // 80 * stride 8

// ---------------- workspace layout (units of 4 bytes unless noted) ----------
constexpr size_t OFF_PBOX = 0;                              // BA*4 f32 (grid units)
constexpr size_t OFF_ALIGN = OFF_PBOX + (size_t)BA * 4;     // BMA f32
constexpr size_t OFF_OVL   = OFF_ALIGN + (size_t)BMA;       // BMA f32
constexpr size_t OFF_TBOX  = OFF_OVL + (size_t)BMA;         // BA*4 f32 (grid units)
constexpr size_t OFF_TSV   = OFF_TBOX + (size_t)BA * 4;     // BA f32
constexpr size_t OFF_PAM   = OFF_TSV + (size_t)BA;          // BM f32
constexpr size_t OFF_POV   = OFF_PAM + BM;                  // BM f32
constexpr size_t OFF_PART  = OFF_POV + BM;                  // 5*PPAD f32
constexpr size_t OFF_TGI   = OFF_PART + 5 * PPAD;           // BA i32
constexpr size_t OFF_FG    = OFF_TGI + BA;                  // BA i32
constexpr size_t OFF_TLAB  = OFF_FG + BA;                   // BA i32
constexpr size_t OFF_MASK_BYTES = (OFF_TLAB + BA) * 4;      // BMA u8 (byte offset)

// ---------------- helpers ----------------
DEV void anchor_of(int a, float& gx, float& gy, float& s) {
  if (a < A0)            { int x = a % 80, y = a / 80;                 gx = x + 0.5f; gy = y + 0.5f; s = 8.0f;  }
  else if (a < A0 + A1)  { int r = a - A0; int x = r % 40, y = r / 40; gx = x + 0.5f; gy = y + 0.5f; s = 16.0f; }
  else                   { int r = a - A0 - A1; int x = r % 20, y = r / 20; gx = x + 0.5f; gy = y + 0.5f; s = 32.0f; }
}

DEV float featAt(const float* f0, const float* f1, const float* f2, int b, int c, int a) {
  if (a < A0)      return f0[((size_t)b * 144 + c) * A0 + a];
  if (a < A0 + A1) return f1[((size_t)b * 144 + c) * A1 + (a - A0)];
  return f2[((size_t)b * 144 + c) * A2v + (a - A0 - A1)];
}

DEV void gt_corners(const float* gtb, int b, int m,
                    float& x1, float& y1, float& x2, float& y2, bool& valid) {
  const float* p = gtb + ((size_t)b * M + m) * 4;
  float cx = p[0] * INPUT, cy = p[1] * INPUT, w = p[2] * INPUT, h = p[3] * INPUT;
  x1 = cx - 0.5f * w; y1 = cy - 0.5f * h; x2 = cx + 0.5f * w; y2 = cy + 0.5f * h;
  valid = (x1 + y1 + x2 + y2) > 0.0f;
}

DEV float ciou_f(float b1x1, float b1y1, float b1x2, float b1y2,
                 float b2x1, float b2y1, float b2x2, float b2y2) {
  const float eps = 1e-7f;
  float w1 = b1x2 - b1x1, h1 = b1y2 - b1y1 + eps;
  float w2 = b2x2 - b2x1, h2 = b2y2 - b2y1 + eps;
  float iw = fmaxf(fminf(b1x2, b2x2) - fmaxf(b1x1, b2x1), 0.0f);
  float ih = fmaxf(fminf(b1y2, b2y2) - fmaxf(b1y1, b2y1), 0.0f);
  float inter = iw * ih;
  float uni = w1 * h1 + w2 * h2 - inter + eps;
  float iou = inter / uni;
  float cw = fmaxf(b1x2, b2x2) - fminf(b1x1, b2x1);
  float ch = fmaxf(b1y2, b2y2) - fminf(b1y1, b2y1);
  float c2 = cw * cw + ch * ch + eps;
  float dx = b2x1 + b2x2 - b1x1 - b1x2;
  float dy = b2y1 + b2y2 - b1y1 - b1y2;
  float rho2 = (dx * dx + dy * dy) * 0.25f;
  float dat = atanf(w2 / h2) - atanf(w1 / h1);
  float v = 0.405284734569351f * dat * dat;         // 4/pi^2
  float alpha = v / (v - iou + (1.0f + eps));
  return iou - (rho2 / c2 + v * alpha);
}

DEV float blockSum256(float v, float* sh) {
  int t = threadIdx.x;
  sh[t] = v; __syncthreads();
  for (int off = 128; off > 0; off >>= 1) {
    if (t < off) sh[t] += sh[t + off];
    __syncthreads();
  }
  float r = sh[0]; __syncthreads();
  return r;
}

// ---------------- K0: zero partial-sum region ----------------
__global__ void k_zero(float* part) {
  int i = blockIdx.x * 256 + threadIdx.x;
  if (i < 5 * PPAD) part[i] = 0.0f;
}

// ---------------- K1: bbox decode (DFL softmax proj) + BCE base partials ----
__global__ void k_prep(const float* f0, const float* f1, const float* f2,
                       float* pbox, float* p_bce) {
  __shared__ float sh[256];
  int idx = blockIdx.x * 256 + threadIdx.x;
  int b = idx / A, a = idx % A;
  float gx, gy, s; anchor_of(a, gx, gy, s);

  float d[4];
  #pragma unroll
  for (int g = 0; g < 4; ++g) {
    float xv[16], mx = -1e30f;
    #pragma unroll
    for (int k = 0; k < 16; ++k) { xv[k] = featAt(f0, f1, f2, b, g * 16 + k, a); mx = fmaxf(mx, xv[k]); }
    float se = 0.0f, num = 0.0f;
    #pragma unroll
    for (int k = 0; k < 16; ++k) { float e = expf(xv[k] - mx); se += e; num += e * (float)k; }
    d[g] = num / se;
  }
  float* pb = pbox + (size_t)idx * 4;
  pb[0] = gx - d[0]; pb[1] = gy - d[1]; pb[2] = gx + d[2]; pb[3] = gy + d[3];

  // BCE base: sum_c max(p,0) + log1p(exp(-|p|))
  float acc = 0.0f;
  for (int c = 64; c < 144; ++c) {
    float p = featAt(f0, f1, f2, b, c, a);
    acc += fmaxf(p, 0.0f) + log1pf(expf(-fabsf(p)));
  }
  float tot = blockSum256(acc, sh);
  if (threadIdx.x == 0) p_bce[blockIdx.x] = tot;
}

// ---------------- K2: pairwise align metric & overlaps ----------------
__global__ void k_pair(const float* f0, const float* f1, const float* f2,
                       const float* gtb, const int* glab, const float* pbox,
                       float* align, float* ovl) {
  int bm = blockIdx.x;
  int b = bm / M, m = bm % M;
  float gx1, gy1, gx2, gy2; bool valid;
  gt_corners(gtb, b, m, gx1, gy1, gx2, gy2, valid);
  float* arow = align + (size_t)bm * A;
  float* orow = ovl + (size_t)bm * A;
  if (!valid) {
    for (int a = threadIdx.x; a < A; a += 256) { arow[a] = 0.0f; orow[a] = 0.0f; }
    return;
  }
  int lab = min(max(glab[b * M + m], 0), NC - 1);
  for (int a = threadIdx.x; a < A; a += 256) {
    int ap = a + 256 < A ? a + 256 : a;
    __builtin_prefetch(pbox + (size_t)(b * A + ap) * 4, 0, 1);   // global_prefetch_b8
    float ax, ay, s; anchor_of(a, ax, ay, s);
    ax *= s; ay *= s;
    float dmin = fminf(fminf(ax - gx1, ay - gy1), fminf(gx2 - ax, gy2 - ay));
    float al = 0.0f, ov = 0.0f;
    if (dmin > EPSA) {
      const float* pb = pbox + (size_t)(b * A + a) * 4;
      float px1 = pb[0] * s, py1 = pb[1] * s, px2 = pb[2] * s, py2 = pb[3] * s;
      float ci = ciou_f(gx1, gy1, gx2, gy2, px1, py1, px2, py2);
      ov = fmaxf(ci, 0.0f);
      float sc = featAt(f0, f1, f2, b, 64 + lab, a);
      float bs = 1.0f / (1.0f + expf(-sc));
      float o3 = ov * ov * ov;
      al = sqrtf(bs) * o3 * o3;        // bs^0.5 * ov^6
    }
    arow[a] = al; orow[a] = ov;
  }
}

// ---------------- K3: stable top-10 per (b,m) out of LDS-staged row ---------
// Row is staged Global->LDS with CDNA5 async-copy (ASYNCcnt), then 10
// selection passes run from LDS; chosen entries are knocked out with a -1
// sentinel (all real align values are >= 0), preserving stable ties.
__global__ void k_topk(const float* gtb, const float* align, unsigned char* mask) {
  __shared__ float sa[A];          // 33600 B staged align row
  __shared__ float sv[256];
  __shared__ int   si[256];
  __shared__ int   chosen[10];
  int bm = blockIdx.x;
  int b = bm / M, m = bm % M;
  float gx1, gy1, gx2, gy2; bool valid;
  gt_corners(gtb, b, m, gx1, gy1, gx2, gy2, valid);
  unsigned char* mrow = mask + (size_t)bm * A;
  for (int i = threadIdx.x; i < A; i += 256) mrow[i] = 0;
  __syncthreads();
  if (!valid) return;

  const float* arow = align + (size_t)bm * A;
  // ---- async Global -> LDS staging ----
  for (int i = threadIdx.x; i < A; i += 256) {
    uint32_t ldsOff = (uint32_t)(uintptr_t)(&sa[i]);   // low 32 bits = LDS byte addr
    uint32_t gOff   = (uint32_t)(i * 4);               // byte offset from row base
    asm volatile("global_load_async_to_lds_b32 %0, %1, %2"
                 :: "v"(ldsOff), "v"(gOff), "s"(arow)
                 : "memory");
  }
  asm volatile("s_wait_asynccnt 0" ::: "memory");       // this wave's copies done
  __syncthreads();                                      // all waves' copies done

  for (int k = 0; k < 10; ++k) {
    float bv = -0.5f; int bi = 1 << 30;
    for (int i = threadIdx.x; i < A; i += 256) {
      float v = sa[i];
      if (v > bv || (v == bv && i < bi)) { bv = v; bi = i; }
    }
    sv[threadIdx.x] = bv; si[threadIdx.x] = bi;
    __syncthreads();
    for (int off = 128; off > 0; off >>= 1) {
      if (threadIdx.x < off) {
        float ov_ = sv[threadIdx.x + off]; int oi = si[threadIdx.x + off];
        if (ov_ > sv[threadIdx.x] || (ov_ == sv[threadIdx.x] && oi < si[threadIdx.x])) {
          sv[threadIdx.x] = ov_; si[threadIdx.x] = oi;
        }
      }
      __syncthreads();
    }
    if (threadIdx.x == 0) { chosen[k] = si[0]; sa[si[0]] = -1.0f; }  // knock out
    __syncthreads();
  }
  if (threadIdx.x < 10) {
    int ai = chosen[threadIdx.x];
    float ax, ay, s; anchor_of(ai, ax, ay, s);
    ax *= s; ay *= s;
    float dmin = fminf(fminf(ax - gx1, ay - gy1), fminf(gx2 - ax, gy2 - ay));
    if (dmin > EPSA) mrow[ai] = 1;   // top-k AND in-gt AND valid
  }
}

// ---------------- K4: resolve multi-GT anchors, build targets ----------------
__global__ void k_resolve(const float* gtb, const int* glab, const float* ovl,
                          unsigned char* mask, int* tgi, int* fgf, int* tlab, float* tbox) {
  int idx = blockIdx.x * 256 + threadIdx.x;
  int b = idx / A, a = idx % A;
  int fg = 0;
  for (int m = 0; m < M; ++m) fg += mask[((size_t)(b * M + m)) * A + a];
  if (fg > 1) {
    float best = ovl[((size_t)(b * M)) * A + a]; int bi = 0;
    for (int m = 1; m < M; ++m) {
      float v = ovl[((size_t)(b * M + m)) * A + a];
      if (v > best) { best = v; bi = m; }
    }
    for (int m = 0; m < M; ++m) mask[((size_t)(b * M + m)) * A + a] = (m == bi) ? 1 : 0;
    fg = 1;
  }
  int t = 0; bool found = false;
  for (int m = 0; m < M; ++m) {
    if (!found && mask[((size_t)(b * M + m)) * A + a]) { t = m; found = true; }
  }
  tgi[idx] = t;
  fgf[idx] = (fg > 0) ? 1 : 0;
  tlab[idx] = max(glab[b * M + t], 0);
  float gx, gy, s; anchor_of(a, gx, gy, s);
  float x1, y1, x2, y2; bool vv;
  gt_corners(gtb, b, t, x1, y1, x2, y2, vv);
  float inv = 1.0f / s;
  float* tb = tbox + (size_t)idx * 4;
  tb[0] = x1 * inv; tb[1] = y1 * inv; tb[2] = x2 * inv; tb[3] = y2 * inv;
}

// ---------------- K5: per-(b,m) pos_am / pos_ov ----------------
__global__ void k_rowmax(const float* align, const float* ovl, const unsigned char* mask,
                         float* pam, float* pov) {
  __shared__ float s1[256], s2[256];
  int bm = blockIdx.x;
  const float* arow = align + (size_t)bm * A;
  const float* orow = ovl + (size_t)bm * A;
  const unsigned char* mrow = mask + (size_t)bm * A;
  float m1 = 0.0f, m2 = 0.0f;
  for (int a = threadIdx.x; a < A; a += 256) {
    if (mrow[a]) { m1 = fmaxf(m1, arow[a]); m2 = fmaxf(m2, orow[a]); }
  }
  s1[threadIdx.x] = m1; s2[threadIdx.x] = m2;
  __syncthreads();
  for (int off = 128; off > 0; off >>= 1) {
    if (threadIdx.x < off) {
      s1[threadIdx.x] = fmaxf(s1[threadIdx.x], s1[threadIdx.x + off]);
      s2[threadIdx.x] = fmaxf(s2[threadIdx.x], s2[threadIdx.x + off]);
    }
    __syncthreads();
  }
  if (threadIdx.x == 0) { pam[bm] = s1[0]; pov[bm] = s2[0]; }
}

// ---------------- K6: per-anchor normalized target score + sum partials -----
__global__ void k_norm(const float* align, const unsigned char* mask,
                       const float* pam, const float* pov, const int* fgf,
                       float* tsv, float* p_ts) {
  __shared__ float sh[256];
  int idx = blockIdx.x * 256 + threadIdx.x;
  int b = idx / A, a = idx % A;
  float nrm = 0.0f;
  for (int m = 0; m < M; ++m) {
    int bm = b * M + m;
    if (mask[((size_t)bm) * A + a]) {
      float t = align[((size_t)bm) * A + a] * pov[bm] / (pam[bm] + EPSA);
      nrm = fmaxf(nrm, t);
    }
  }
  float ts = fgf[idx] ? nrm : 0.0f;
  tsv[idx] = ts;
  float tot = blockSum256(ts, sh);
  if (threadIdx.x == 0) p_ts[blockIdx.x] = tot;
}

// ---------------- K7: per-anchor box / cls-correction / DFL partials --------
__global__ void k_loss(const float* f0, const float* f1, const float* f2,
                       const float* pbox, const float* tbox, const float* tsv,
                       const int* fgf, const int* tlab,
                       float* p_box, float* p_cls, float* p_dfl) {
  __shared__ float sh[256];
  int idx = blockIdx.x * 256 + threadIdx.x;
  int b = idx / A, a = idx % A;
  float lb = 0.0f, lc = 0.0f, ld = 0.0f;
  if (fgf[idx]) {
    float w = tsv[idx];
    const float* pb = pbox + (size_t)idx * 4;
    const float* tb = tbox + (size_t)idx * 4;
    float iou = ciou_f(pb[0], pb[1], pb[2], pb[3], tb[0], tb[1], tb[2], tb[3]);
    lb = (1.0f - iou) * w;
    lc = featAt(f0, f1, f2, b, 64 + tlab[idx], a) * w;     // sum p*t correction
    float gx, gy, s; anchor_of(a, gx, gy, s);
    float tgt[4] = { gx - tb[0], gy - tb[1], tb[2] - gx, tb[3] - gy };
    float dsum = 0.0f;
    #pragma unroll
    for (int g = 0; g < 4; ++g) {
      float tg = fminf(fmaxf(tgt[g], 0.0f), 14.99f);
      float xv[16], mx = -1e30f;
      #pragma unroll
      for (int k = 0; k < 16; ++k) { xv[k] = featAt(f0, f1, f2, b, g * 16 + k, a); mx = fmaxf(mx, xv[k]); }
      float se = 0.0f;
      #pragma unroll
      for (int k = 0; k < 16; ++k) se += expf(xv[k] - mx);
      float lse = mx + logf(se);
      int tl = (int)tg;
      float wl = (float)(tl + 1) - tg;
      float ll = -(xv[tl] - lse);
      float lr = -(xv[tl + 1] - lse);
      dsum += ll * wl + lr * (1.0f - wl);
    }
    ld = dsum * 0.25f * w;
  }
  float t0 = blockSum256(lb, sh);
  float t1 = blockSum256(lc, sh);
  float t2 = blockSum256(ld, sh);
  if (threadIdx.x == 0) { p_box[blockIdx.x] = t0; p_cls[blockIdx.x] = t1; p_dfl[blockIdx.x] = t2; }
}

// ---------------- K8: WMMA f32 partial-sum reduction + final losses ----------
typedef float v2f __attribute__((ext_vector_type(2)));
typedef float v8f __attribute__((ext_vector_type(8)));

DEV float wmma_sum576(const float* p) {
  // A-matrix 16x4 f32 layout: lane L(<16) holds A[L][0..1]; lane L+16 holds A[L][2..3]
  v8f acc = {0.f, 0.f, 0.f, 0.f, 0.f, 0.f, 0.f, 0.f};
  v2f bone; bone.x = 1.0f; bone.y = 1.0f;   // B = ones(4x16)
  int lane = threadIdx.x;
  int base0 = (lane & 15) * 4 + ((lane >> 4) << 1);
  for (int t = 0; t < 9; ++t) {
    const float* q = p + t * 64 + base0;
    v2f av; av.x = q[0]; av.y = q[1];
    acc = __builtin_amdgcn_wmma_f32_16x16x4_f32(false, av, false, bone,
                                                (short)0, acc, false, false);
  }
  float s = 0.0f;
  #pragma unroll
  for (int r = 0; r < 8; ++r) s += acc[r];  // rows (M half) of column `lane`
  s += __shfl_xor(s, 16, 32);               // add the other 8 rows
  return s;                                  // total, uniform across lanes
}

__global__ void k_final(const float* part, float* out) {
  float Sbce = wmma_sum576(part + 0 * PPAD);
  float Sts  = wmma_sum576(part + 1 * PPAD);
  float Sbox = wmma_sum576(part + 2 * PPAD);
  float Scls = wmma_sum576(part + 3 * PPAD);
  float Sdfl = wmma_sum576(part + 4 * PPAD);
  if (threadIdx.x == 0) {
    float tss = fmaxf(Sts, 1.0f);
    out[0] = 7.5f * Sbox / tss;                 // loss_box * gain
    out[1] = 0.5f * (Sbce - Scls) / tss;        // loss_cls * gain
    out[2] = 1.5f * Sdfl / tss;                 // loss_dfl * gain
  }
}

// ---------------- host launcher ----------------
extern "C" void kernel_launch(void* const* d_in, const int* in_sizes, int n_in,
                              void* d_out, int out_size, void* d_ws, size_t ws_size,
                              hipStream_t stream) {
  (void)in_sizes; (void)n_in; (void)out_size; (void)ws_size;
  const float* f0   = (const float*)d_in[0];
  const float* f1   = (const float*)d_in[1];
  const float* f2   = (const float*)d_in[2];
  const int*   glab = (const int*)d_in[3];
  const float* gtb  = (const float*)d_in[4];
  float* out = (float*)d_out;

  float* ws = (float*)d_ws;
  float* pbox  = ws + OFF_PBOX;
  float* align = ws + OFF_ALIGN;
  float* ovl   = ws + OFF_OVL;
  float* tbox  = ws + OFF_TBOX;
  float* tsv   = ws + OFF_TSV;
  float* pam   = ws + OFF_PAM;
  float* pov   = ws + OFF_POV;
  float* part  = ws + OFF_PART;
  int*   tgi   = (int*)ws + OFF_TGI;
  int*   fgf   = (int*)ws + OFF_FG;
  int*   tlab  = (int*)ws + OFF_TLAB;
  unsigned char* mask = (unsigned char*)d_ws + OFF_MASK_BYTES;

  float* p_bce = part + 0 * PPAD;
  float* p_ts  = part + 1 * PPAD;
  float* p_box = part + 2 * PPAD;
  float* p_cls = part + 3 * PPAD;
  float* p_dfl = part + 4 * PPAD;

  k_zero   <<<(5 * PPAD + 255) / 256, 256, 0, stream>>>(part);
  k_prep   <<<NBLK, 256, 0, stream>>>(f0, f1, f2, pbox, p_bce);
  k_pair   <<<BM,   256, 0, stream>>>(f0, f1, f2, gtb, glab, pbox, align, ovl);
  k_topk   <<<BM,   256, 0, stream>>>(gtb, align, mask);
  k_resolve<<<NBLK, 256, 0, stream>>>(gtb, glab, ovl, mask, tgi, fgf, tlab, tbox);
  k_rowmax <<<BM,   256, 0, stream>>>(align, ovl, mask, pam, pov);
  k_norm   <<<NBLK, 256, 0, stream>>>(align, mask, pam, pov, fgf, tsv, p_ts);
  k_loss   <<<NBLK, 256, 0, stream>>>(f0, f1, f2, pbox, tbox, tsv, fgf, tlab,
                                      p_box, p_cls, p_dfl);
  k_final  <<<1, 32, 0, stream>>>(part, out);
}